// SolidLinear_70832600646263
// MI455X (gfx1250) — compile-verified
//
#include <hip/hip_runtime.h>

// Tropical (min-plus + max-plus) "linear layer" for MI455X / gfx1250.
// out[b,o] = min_i(W[o,i]+X[b,i]) + max_i(W[o,i]+X[b,i])
// B=1024, OUT=1024, IN=512, all f32.
//
// VALU-bound problem (no ring matmul -> WMMA inapplicable). Strategy:
//  * 64x64 output tile per 256-thread workgroup, 4x4 outputs per thread.
//  * K chunked by 64; chunks staged TRANSPOSED and K-PAIR-INTERLEAVED into
//    LDS via global_load_async_to_lds_b32 (ASYNCcnt tracked), double-buffered
//    and overlapped with compute.
//  * Inner loop processes k in pairs, with forced CDNA5 ALU ops:
//    v_pk_add_f32 (both sums in one op), v_min3_num_f32 / v_max3_num_f32
//    (fold both sums into the accumulator in one op each):
//    48 VALU per 2 k-steps per thread instead of 96.
// (Kernel renamed this round to verify a fresh compile in the disasm.)

#define B_DIM   1024
#define OUT_DIM 1024
#define IN_DIM  512

#define BM 64           // batch rows per tile
#define BO 64           // output cols per tile
#define BK 64           // K chunk
#define NCHUNK (IN_DIM / BK)
#define NTHREADS 256

typedef float v2f __attribute__((ext_vector_type(2)));

// ---- CDNA5 async load: memory -> LDS, 4 bytes per lane, GVS addressing ----
__device__ __forceinline__ void async_ld_b32(unsigned lds_addr, unsigned voff,
                                             const float* sbase) {
  asm volatile("global_load_async_to_lds_b32 %0, %1, %2"
               :: "v"(lds_addr), "v"(voff), "s"(sbase)
               : "memory");
}

__device__ __forceinline__ void wait_async0() {
#if __has_builtin(__builtin_amdgcn_s_wait_asynccnt)
  __builtin_amdgcn_s_wait_asynccnt(0);
#else
  asm volatile("s_wait_asynccnt 0" ::: "memory");
#endif
}

// ---- Forced CDNA5 VALU ops (backend fails to pattern-match these) ----
__device__ __forceinline__ v2f pk_add_f32(v2f a, v2f b) {
  v2f d;
  asm("v_pk_add_f32 %0, %1, %2" : "=v"(d) : "v"(a), "v"(b));
  return d;
}
__device__ __forceinline__ float min3_f32(float a, float b, float c) {
  float d;
  asm("v_min3_num_f32 %0, %1, %2, %3" : "=v"(d) : "v"(a), "v"(b), "v"(c));
  return d;
}
__device__ __forceinline__ float max3_f32(float a, float b, float c) {
  float d;
  asm("v_max3_num_f32 %0, %1, %2, %3" : "=v"(d) : "v"(a), "v"(b), "v"(c));
  return d;
}

// Generic pointer to a __shared__ object: low 32 bits are the LDS offset
// (LDS aperture is 4GB-aligned).
__device__ __forceinline__ unsigned lds_off(const void* p) {
  return (unsigned)(unsigned long long)p;
}

__global__ __launch_bounds__(NTHREADS)
void tropical_minmax3_kernel(const float* __restrict__ X,
                             const float* __restrict__ W,
                             float* __restrict__ out) {
  // Pair-interleaved transposed staging:
  //   Xs[buf][(k>>1)*(2*BM) + b*2 + (k&1)]
  //   Ws[buf][(k>>1)*(2*BO) + o*2 + (k&1)]
  // so the two k-values of a pair sit in adjacent floats per b/o -> v2f regs.
  __shared__ float Xs[2][BK * BM];   // 2 x 16 KB
  __shared__ float Ws[2][BK * BO];   // 2 x 16 KB

  const int tid = threadIdx.x;
  const int tx  = tid & 15;          // o-block index (0..15)
  const int ty  = tid >> 4;          // b-block index (0..15)
  const int bBase = blockIdx.y * BM;
  const int oBase = blockIdx.x * BO;

  const unsigned xsBase = lds_off(&Xs[0][0]);
  const unsigned wsBase = lds_off(&Ws[0][0]);

  // Per-lane piece of the chunk fill. e = tid + it*256; row = e>>6, k = e&63.
  // Consecutive lanes -> consecutive k -> coalesced 128B global segments.
  // LDS dest is the pair-interleaved transposed slot.
  auto load_chunk = [&](int buf, int k0) {
    const float* xb = X + (size_t)bBase * IN_DIM + k0;   // uniform -> SGPRs
    const float* wb = W + (size_t)oBase * IN_DIM + k0;   // uniform -> SGPRs
#pragma unroll
    for (int it = 0; it < (BK * BM) / NTHREADS; ++it) {  // 16 iters
      const int e   = tid + it * NTHREADS;
      const int row = e >> 6;        // b (for X) or o (for W)
      const int k   = e & 63;
      const unsigned goff = (unsigned)((row * IN_DIM + k) * 4);
      const int slot = (k >> 1) * 128 + row * 2 + (k & 1);
      const unsigned lx = xsBase + (unsigned)((buf * BK * BM + slot) * 4);
      const unsigned lw = wsBase + (unsigned)((buf * BK * BO + slot) * 4);
      async_ld_b32(lx, goff, xb);
      async_ld_b32(lw, goff, wb);
    }
  };

  float mn[16], mx[16];
#pragma unroll
  for (int i = 0; i < 16; ++i) { mn[i] = INFINITY; mx[i] = -INFINITY; }

  load_chunk(0, 0);

  for (int c = 0; c < NCHUNK; ++c) {
    wait_async0();       // this wave's portion of chunk c has landed in LDS
    __syncthreads();     // everyone's portion landed; prev compute finished

    if (c + 1 < NCHUNK)  // prefetch next chunk into the other buffer,
      load_chunk((c + 1) & 1, (c + 1) * BK);  // overlapped with compute below

    const float* xs = &Xs[c & 1][0];
    const float* ws = &Ws[c & 1][0];

#pragma unroll 2
    for (int k2 = 0; k2 < BK / 2; ++k2) {     // two k-values per iteration
      const v2f* xp = (const v2f*)(xs + k2 * (2 * BM)) + ty * 4;
      const v2f* wp = (const v2f*)(ws + k2 * (2 * BO)) + tx * 4;
      v2f xa[4], wa[4];
#pragma unroll
      for (int i = 0; i < 4; ++i) xa[i] = xp[i];   // merges into ds_load_b128
#pragma unroll
      for (int j = 0; j < 4; ++j) wa[j] = wp[j];
#pragma unroll
      for (int i = 0; i < 4; ++i) {
#pragma unroll
        for (int j = 0; j < 4; ++j) {
          const v2f s = pk_add_f32(xa[i], wa[j]);   // v_pk_add_f32
          const int a = i * 4 + j;
          mn[a] = min3_f32(mn[a], s.x, s.y);        // v_min3_num_f32
          mx[a] = max3_f32(mx[a], s.x, s.y);        // v_max3_num_f32
        }
      }
    }
  }

  // Epilogue: out[b,o] = min + max, 4x global_store_b128 per thread.
  const int b0 = bBase + ty * 4;
  const int o0 = oBase + tx * 4;
#pragma unroll
  for (int i = 0; i < 4; ++i) {
    float4 r;
    r.x = mn[i * 4 + 0] + mx[i * 4 + 0];
    r.y = mn[i * 4 + 1] + mx[i * 4 + 1];
    r.z = mn[i * 4 + 2] + mx[i * 4 + 2];
    r.w = mn[i * 4 + 3] + mx[i * 4 + 3];
    *(float4*)(out + (size_t)(b0 + i) * OUT_DIM + o0) = r;
  }
}

extern "C" void kernel_launch(void* const* d_in, const int* in_sizes, int n_in,
                              void* d_out, int out_size, void* d_ws, size_t ws_size,
                              hipStream_t stream) {
  (void)in_sizes; (void)n_in; (void)out_size; (void)d_ws; (void)ws_size;
  const float* X = (const float*)d_in[0];   // [B, IN]
  const float* W = (const float*)d_in[1];   // [OUT, IN]
  float* out = (float*)d_out;               // [B, OUT]

  dim3 grid(OUT_DIM / BO, B_DIM / BM);      // 16 x 16
  tropical_minmax3_kernel<<<grid, NTHREADS, 0, stream>>>(X, W, out);
}